// QueryProvidedAttentionBlock_19353122635976
// MI455X (gfx1250) — compile-verified
//
#include <hip/hip_runtime.h>
#include <cstdint>
#include <cstddef>

// Problem constants (B,T,C,G from the reference)
#define B_ 8
#define T_ 8192
#define C_ 1024
#define G_ 32
#define CG_ 32
#define EPS_ 1e-5f
#define SCALE2_ (1.0f/32.0f)   // SCALE^2 = 1/sqrt(C)

typedef __attribute__((ext_vector_type(2))) float v2f;
typedef __attribute__((ext_vector_type(8))) float v8f;
typedef __attribute__((ext_vector_type(4))) unsigned int v4u;
typedef __attribute__((ext_vector_type(4))) int v4i;
typedef __attribute__((ext_vector_type(8))) int v8i;

#if defined(__HIP_DEVICE_COMPILE__) && __has_builtin(__builtin_amdgcn_tensor_load_to_lds) && __has_builtin(__builtin_amdgcn_s_wait_tensorcnt)
#define USE_TDM 1
#else
#define USE_TDM 0
#endif

// ---------------------------------------------------------------------------
// Pass 1: GroupNorm statistics over kvx, per (b,g), reduced over (T, C/G).
// Two-stage deterministic reduction (no float atomics).
// ---------------------------------------------------------------------------
__global__ void k_gn_part(const float* __restrict__ kvx,
                          float* __restrict__ psum, float* __restrict__ psq) {
  const int NS = 8, TS = T_ / NS;        // 8 T-slices of 1024 rows
  int blk = blockIdx.x;                  // blk = bg*8 + s
  int s   = blk & 7;
  int bg  = blk >> 3;
  int b   = bg >> 5, g = bg & 31;
  int tid = threadIdx.x;
  int cc  = tid & 31, tr = tid >> 5;     // 32 channels x 8 rows per sweep
  const float* base = kvx + ((size_t)b * T_ + (size_t)s * TS) * C_ + g * CG_ + cc;
  float sum = 0.f, sq = 0.f;
  for (int t = tr; t < TS; t += 8) {
    float x = base[(size_t)t * C_];
    sum += x; sq += x * x;
  }
  __shared__ float ls[256], lq[256];
  ls[tid] = sum; lq[tid] = sq; __syncthreads();
  for (int off = 128; off > 0; off >>= 1) {
    if (tid < off) { ls[tid] += ls[tid + off]; lq[tid] += lq[tid + off]; }
    __syncthreads();
  }
  if (tid == 0) { psum[blk] = ls[0]; psq[blk] = lq[0]; }
}

// Finalize stats -> per-(b,c) fused affine:  xn = raw*sc[b,c] + of[b,c]
__global__ void k_gn_final(const float* __restrict__ psum, const float* __restrict__ psq,
                           const float* __restrict__ gn_g, const float* __restrict__ gn_b,
                           float* __restrict__ sc, float* __restrict__ of) {
  int bg = threadIdx.x;                  // 256 groups, 1 block
  int b = bg >> 5, g = bg & 31;
  float s = 0.f, q = 0.f;
  for (int i = 0; i < 8; i++) { s += psum[bg * 8 + i]; q += psq[bg * 8 + i]; }
  const float N = (float)(T_ * CG_);
  float mu = s / N;
  float var = q / N - mu * mu;
  float rstd = rsqrtf(var + EPS_);
  for (int j = 0; j < CG_; j++) {
    int c = g * CG_ + j;
    float gg = rstd * gn_g[c];
    sc[b * C_ + c] = gg;
    of[b * C_ + c] = gn_b[c] - mu * gg;
  }
}

// ---------------------------------------------------------------------------
// LayerNorm of qx (8 rows of 1024)
// ---------------------------------------------------------------------------
__global__ void k_ln_q(const float* __restrict__ qx, const float* __restrict__ ln_g,
                       const float* __restrict__ ln_b, float* __restrict__ qn) {
  int b = blockIdx.x, tid = threadIdx.x;
  const float* x = qx + b * C_;
  float s = 0.f, q = 0.f;
  for (int c = tid; c < C_; c += 256) { float v = x[c]; s += v; q += v * v; }
  __shared__ float ls[256], lq[256];
  ls[tid] = s; lq[tid] = q; __syncthreads();
  for (int off = 128; off > 0; off >>= 1) {
    if (tid < off) { ls[tid] += ls[tid + off]; lq[tid] += lq[tid + off]; }
    __syncthreads();
  }
  float mu = ls[0] / C_;
  float var = lq[0] / C_ - mu * mu;
  float rstd = rsqrtf(var + EPS_);
  for (int c = tid; c < C_; c += 256)
    qn[b * C_ + c] = (x[c] - mu) * rstd * ln_g[c] + ln_b[c];
}

// ---------------------------------------------------------------------------
// Generic tiny batched GEMV: Y[b,n] = alpha * sum_k X[b,k]*W[n*sN + k*sK]
//                                    (+ bias[n]) (+ bscale[b]*bias2[n])
// M=8 is tiny -> plain fp32 VALU, X cached in LDS, W streamed once.
// ---------------------------------------------------------------------------
__global__ void k_gemv8(const float* __restrict__ X, const float* __restrict__ W,
                        long sN, long sK,
                        const float* __restrict__ bias, const float* __restrict__ bias2,
                        const float* __restrict__ bscale, float alpha,
                        float* __restrict__ Y, int N) {
  __shared__ float Xl[B_ * C_];
  int tid = threadIdx.x;
  for (int j = 0; j < B_ * C_ / 256; j++) Xl[tid + 256 * j] = X[tid + 256 * j];
  __syncthreads();
  int n = blockIdx.x * 256 + tid;
  if (n >= N) return;
  float acc[B_];
#pragma unroll
  for (int b = 0; b < B_; b++) acc[b] = 0.f;
  const float* Wp = W + (long)n * sN;
  for (int k = 0; k < C_; k++) {
    float wv = Wp[(long)k * sK];
#pragma unroll
    for (int b = 0; b < B_; b++) acc[b] += Xl[b * C_ + k] * wv;
  }
  for (int b = 0; b < B_; b++) {
    float y = alpha * acc[b];
    if (bias)  y += bias[n];
    if (bias2) y += bscale[b] * bias2[n];
    Y[b * N + n] = y;
  }
}

// logit_const[b] = SCALE^2 * dot(q[b], kvb[:C])
__global__ void k_dot8(const float* __restrict__ q, const float* __restrict__ kvb,
                       float* __restrict__ cst) {
  int tid = threadIdx.x;
  int b = tid >> 5, l = tid & 31;
  float s = 0.f;
  for (int j = l; j < C_; j += 32) s += q[b * C_ + j] * kvb[j];
  for (int off = 16; off > 0; off >>= 1) s += __shfl_down(s, off, 32);
  if (l == 0) cst[b] = SCALE2_ * s;
}

// ---------------------------------------------------------------------------
// Heavy pass A: logits[b,t] = qk_s[b] . xn[b,t]  + const_b
// Block = 256 thr (8 waves), 128 t-rows, C chunked by 64.
// kvx tile staged to LDS via TDM (row stride padded 64->68 DW by TDM pad HW),
// normalized in LDS, then V_WMMA_F32_16X16X4_F32 with A = xn rows (t on M),
// B = qk vector in column N=0.  D[:,0] = logits.
// ---------------------------------------------------------------------------
#define TPAD 68
#define CK   64

__global__ void k_logits(const float* __restrict__ kvx, const float* __restrict__ qk_s,
                         const float* __restrict__ scg, const float* __restrict__ ofg,
                         const float* __restrict__ cstp, float* __restrict__ logits) {
  __shared__ float tile[128 * TPAD];
  __shared__ float qkl[C_];
  __shared__ float scl[CK], ofl[CK];
  int tid  = threadIdx.x;
  int b    = blockIdx.x >> 6;
  int tch  = blockIdx.x & 63;
  int t0   = tch * 128;
  int wid  = tid >> 5, lane = tid & 31;
  const size_t rowbase = ((size_t)b * T_ + t0) * C_;

  for (int j = 0; j < 4; j++) qkl[tid + 256 * j] = qk_s[b * C_ + tid + 256 * j];

  v8f acc = {0.f, 0.f, 0.f, 0.f, 0.f, 0.f, 0.f, 0.f};
  int rsel  = 16 * wid + (lane & 15);
  int khalf = 2 * (lane >> 4);
  bool bm   = (lane & 15) == 0;

  for (int c0 = 0; c0 < C_; c0 += CK) {
    __syncthreads();                        // previous tile fully consumed
#if USE_TDM
    if (tid < 32) {                         // one wave issues the TDM descriptor
      uint64_t ga = (uint64_t)(uintptr_t)(kvx + rowbase + c0);
      uint32_t la = (uint32_t)(uintptr_t)&tile[0];
      v4u g0;
      g0.x = 1u;                                            // count=1 (valid D#)
      g0.y = la;                                            // lds_addr
      g0.z = (uint32_t)ga;                                  // global_addr[31:0]
      g0.w = (uint32_t)((ga >> 32) & 0x1FFFFFFu) | (2u << 30); // addr[56:32]|type=2
      v8i g1;
      g1[0] = 0x07520000;   // data_size=4B, pad_enable, interval=64DW, amount=4DW
      g1[1] = 0;            // tensor_dim0[15:0]=0 (dim0=1<<20)
      g1[2] = 0x00000010;   // tensor_dim0[31:16]=0x10, tensor_dim1[15:0]=0
      g1[3] = 0x00400010;   // tensor_dim1[31:16]=0x10, tile_dim0=64
      g1[4] = 128;          // tile_dim1=128, tile_dim2=0
      g1[5] = C_;           // tensor_dim0_stride = 1024 elements
      g1[6] = 0;
      g1[7] = 0;
      v4i gz4 = {0, 0, 0, 0};
      v8i gz8 = {0, 0, 0, 0, 0, 0, 0, 0};
      __builtin_amdgcn_tensor_load_to_lds(g0, g1, gz4, gz4, gz8, 0);
      __builtin_amdgcn_s_wait_tensorcnt(0);
    }
#else
    for (int j = 0; j < 8; j++) {           // cooperative fallback staging
      int qi = tid + 256 * j;               // 2048 float4 = 128x64 tile
      int r = qi >> 4, cc = (qi & 15) * 4;
      float4 vv = *(const float4*)(kvx + rowbase + (size_t)r * C_ + c0 + cc);
      *(float4*)&tile[r * TPAD + cc] = vv;
    }
#endif
    if (tid < CK)            scl[tid]      = scg[b * C_ + c0 + tid];
    else if (tid < 2 * CK)   ofl[tid - CK] = ofg[b * C_ + c0 + tid - CK];
    __syncthreads();
    // fused GroupNorm affine in LDS: xn = raw*sc[c] + of[c]
    for (int j = 0; j < 32; j++) {
      int idx = tid + 256 * j;              // 8192 elements
      int r = idx >> 6, k = idx & 63;
      int ad = r * TPAD + k;
      tile[ad] = fmaf(tile[ad], scl[k], ofl[k]);
    }
    __syncthreads();
    // 16 WMMAs cover 16 t-rows x 64 K per wave
    for (int k0 = 0; k0 < CK; k0 += 4) {
      v2f a2 = *(const v2f*)&tile[rsel * TPAD + k0 + khalf];
      v2f q2 = *(const v2f*)&qkl[c0 + k0 + khalf];
      v2f b2; b2.x = bm ? q2.x : 0.f; b2.y = bm ? q2.y : 0.f;
      acc = __builtin_amdgcn_wmma_f32_16x16x4_f32(false, a2, false, b2,
                                                  (short)0, acc, false, false);
    }
  }
  float cst = cstp[b];
  if ((lane & 15) == 0) {                   // column N=0 lives in lanes 0 and 16
    int tb = t0 + 16 * wid + 8 * (lane >> 4);
#pragma unroll
    for (int m = 0; m < 8; m++) logits[(size_t)b * T_ + tb + m] = acc[m] + cst;
  }
}

// ---------------------------------------------------------------------------
// Softmax over T per batch, then w = softmax*mask, plus wsum[b] = sum_t w.
// ---------------------------------------------------------------------------
__global__ void k_softmax(const float* __restrict__ logits, const int* __restrict__ mask,
                          float* __restrict__ w, float* __restrict__ wsum) {
  __shared__ float l[T_];
  __shared__ float red[256];
  int b = blockIdx.x, tid = threadIdx.x;
  for (int j = 0; j < 32; j++) l[tid + 256 * j] = logits[(size_t)b * T_ + tid + 256 * j];
  __syncthreads();
  float m = -1e30f;
  for (int j = 0; j < 32; j++) m = fmaxf(m, l[tid + 256 * j]);
  red[tid] = m; __syncthreads();
  for (int off = 128; off > 0; off >>= 1) {
    if (tid < off) red[tid] = fmaxf(red[tid], red[tid + off]);
    __syncthreads();
  }
  m = red[0]; __syncthreads();
  float s = 0.f;
  for (int j = 0; j < 32; j++) {
    float e = __expf(l[tid + 256 * j] - m);
    l[tid + 256 * j] = e; s += e;
  }
  red[tid] = s; __syncthreads();
  for (int off = 128; off > 0; off >>= 1) {
    if (tid < off) red[tid] += red[tid + off];
    __syncthreads();
  }
  float inv = 1.f / red[0]; __syncthreads();
  float wloc = 0.f;
  for (int j = 0; j < 32; j++) {
    int t = tid + 256 * j;
    float wv = l[t] * inv * (float)mask[(size_t)b * T_ + t];
    w[(size_t)b * T_ + t] = wv;
    wloc += wv;
  }
  red[tid] = wloc; __syncthreads();
  for (int off = 128; off > 0; off >>= 1) {
    if (tid < off) red[tid] += red[tid + off];
    __syncthreads();
  }
  if (tid == 0) wsum[b] = red[0];
}

// ---------------------------------------------------------------------------
// Heavy pass B: r[b,c] = sum_t w[b,t]*kvx[b,t,c] via WMMA (A = w in row M=0,
// B = kvx rows read coalesced straight from global). Two-stage deterministic
// partial reduction over 8 T-chunks.
// ---------------------------------------------------------------------------
__global__ void k_r_part(const float* __restrict__ kvx, const float* __restrict__ w,
                         float* __restrict__ rpart) {
  __shared__ float wl[1024];
  int tid = threadIdx.x;
  int blk = blockIdx.x;                 // b*64 + tch*8 + cb
  int cb  = blk & 7;
  int tch = (blk >> 3) & 7;
  int b   = blk >> 6;
  for (int j = 0; j < 4; j++)
    wl[tid + 256 * j] = w[(size_t)b * T_ + tch * 1024 + tid + 256 * j];
  __syncthreads();
  int wid = tid >> 5, lane = tid & 31;
  int c0 = cb * 128 + 16 * wid;
  int khalf = 2 * (lane >> 4);
  bool am = (lane & 15) == 0;
  v8f acc = {0.f, 0.f, 0.f, 0.f, 0.f, 0.f, 0.f, 0.f};
  const float* base = kvx + ((size_t)b * T_ + tch * 1024) * C_ + c0 + (lane & 15);
  for (int tl = 0; tl < 1024; tl += 4) {
    v2f w2 = *(const v2f*)&wl[tl + khalf];
    v2f a2; a2.x = am ? w2.x : 0.f; a2.y = am ? w2.y : 0.f;
    const float* p = base + (size_t)(tl + khalf) * C_;
    v2f b2; b2.x = p[0]; b2.y = p[C_];
    acc = __builtin_amdgcn_wmma_f32_16x16x4_f32(false, a2, false, b2,
                                                (short)0, acc, false, false);
  }
  if (lane < 16)                          // row M=0 of D: VGPR0, lanes 0..15
    rpart[((size_t)b * 8 + tch) * C_ + c0 + lane] = acc[0];
}

// u[b,c] = sc[b,c]*r[b,c] + of[b,c]*wsum[b]
__global__ void k_r_final(const float* __restrict__ rpart, const float* __restrict__ scg,
                          const float* __restrict__ ofg, const float* __restrict__ wsum,
                          float* __restrict__ u) {
  int idx = blockIdx.x * 256 + threadIdx.x;     // 8192
  int b = idx >> 10, c = idx & 1023;
  float r = 0.f;
  for (int ch = 0; ch < 8; ch++) r += rpart[((size_t)b * 8 + ch) * C_ + c];
  u[idx] = scg[idx] * r + ofg[idx] * wsum[b];
}

// out[b,t,c] = kvx[b,t,c] + h[b,c]
__global__ void k_final_add(const float* __restrict__ kvx, const float* __restrict__ h,
                            float* __restrict__ out) {
  size_t i4 = (size_t)blockIdx.x * 256 + threadIdx.x;
  size_t e = i4 * 4;
  int c = (int)(e & (size_t)(C_ - 1));
  int b = (int)(e >> 23);                       // T*C = 2^23
  float4 x  = *(const float4*)(kvx + e);
  float4 hv = *(const float4*)(h + b * C_ + c);
  float4 o; o.x = x.x + hv.x; o.y = x.y + hv.y; o.z = x.z + hv.z; o.w = x.w + hv.w;
  *(float4*)(out + e) = o;
}

// ---------------------------------------------------------------------------
extern "C" void kernel_launch(void* const* d_in, const int* in_sizes, int n_in,
                              void* d_out, int out_size, void* d_ws, size_t ws_size,
                              hipStream_t stream) {
  const float* qx    = (const float*)d_in[0];
  const float* kvx   = (const float*)d_in[1];
  const int*   mask  = (const int*)d_in[2];
  const float* ln_g  = (const float*)d_in[3];
  const float* ln_b  = (const float*)d_in[4];
  const float* qW    = (const float*)d_in[5];
  const float* qb    = (const float*)d_in[6];
  const float* gn_g  = (const float*)d_in[7];
  const float* gn_b  = (const float*)d_in[8];
  const float* kvW   = (const float*)d_in[9];
  const float* kvb   = (const float*)d_in[10];
  const float* projW = (const float*)d_in[11];
  const float* projb = (const float*)d_in[12];
  float* out = (float*)d_out;
  float* ws  = (float*)d_ws;

  float* qn     = ws;             // 8192
  float* q      = ws + 8192;      // 8192
  float* qk     = ws + 16384;     // 8192
  float* u      = ws + 24576;     // 8192
  float* a      = ws + 32768;     // 8192
  float* h      = ws + 40960;     // 8192
  float* sc     = ws + 49152;     // 8192
  float* of     = ws + 57344;     // 8192
  float* psum   = ws + 65536;     // 2048
  float* psq    = ws + 67584;     // 2048
  float* cst    = ws + 69632;     // 8
  float* wsum   = ws + 69648;     // 8
  float* logits = ws + 69664;     // 65536
  float* wbuf   = ws + 135200;    // 65536
  float* rpart  = ws + 200736;    // 65536   (total ~1.07 MB)

  k_gn_part  <<<2048, 256, 0, stream>>>(kvx, psum, psq);
  k_gn_final <<<1,    256, 0, stream>>>(psum, psq, gn_g, gn_b, sc, of);
  k_ln_q     <<<8,    256, 0, stream>>>(qx, ln_g, ln_b, qn);
  k_gemv8    <<<4,    256, 0, stream>>>(qn, qW, (long)C_, 1L, qb, nullptr, nullptr, 1.0f, q, C_);
  k_gemv8    <<<4,    256, 0, stream>>>(q, kvW, 1L, (long)C_, nullptr, nullptr, nullptr, SCALE2_, qk, C_);
  k_dot8     <<<1,    256, 0, stream>>>(q, kvb, cst);
  k_logits   <<<512,  256, 0, stream>>>(kvx, qk, sc, of, cst, logits);
  k_softmax  <<<8,    256, 0, stream>>>(logits, mask, wbuf, wsum);
  k_r_part   <<<512,  256, 0, stream>>>(kvx, wbuf, rpart);
  k_r_final  <<<32,   256, 0, stream>>>(rpart, sc, of, wsum, u);
  k_gemv8    <<<4,    256, 0, stream>>>(u, kvW + (size_t)C_ * C_, (long)C_, 1L, nullptr, kvb + C_, wsum, 1.0f, a, C_);
  k_gemv8    <<<4,    256, 0, stream>>>(a, projW, (long)C_, 1L, projb, nullptr, nullptr, 1.0f, h, C_);
  k_final_add<<<65536, 256, 0, stream>>>(kvx, h, out);
}